// HardInfoNCESyncLoss_34583076667398
// MI455X (gfx1250) — compile-verified
//
#include <hip/hip_runtime.h>
#include <math.h>

// ---------------------------------------------------------------------------
// HardInfoNCE loss, MI455X (gfx1250), wave32 + V_WMMA_F32_16X16X4_F32 path.
//   B = 8192 rows, D = 512 features, temperature = 0.07, top-5 hard negatives.
// Pipeline:
//   1) nrm_k     : L2-normalize rows of v_emb / a_emb into workspace.
//   2) simtopk_k : fused sim-tile GEMM (f32 WMMA) + streaming per-row top-5
//                  (+ diagonal positive) + logsumexp; one row-loss per row.
//                  Two N-tiles per pass -> 2 independent WMMA accumulation
//                  chains per wave, each A fragment reused for 2 WMMAs.
//   3) red_k     : reduce 2*B row losses -> scalar 0.5*(mean+mean).
// ---------------------------------------------------------------------------

typedef float v2f __attribute__((ext_vector_type(2)));
typedef float v8f __attribute__((ext_vector_type(8)));

#define BN 8192
#define DN 512
#define INV_T 14.285714285714286f /* 1/0.07 */
#define KTOP 5
#define ROWS_PER_WG 128           /* 8 waves x 16-row strips              */
#define NTILE 2                   /* column tiles per pass (N = 32)       */
#define BS_STRIDE 516             /* 516 mod 64 == 4 -> conflict-free LDS */

__device__ __forceinline__ float neg_inf() { return -__builtin_inff(); }

// ---------------------------------------------------------------- normalize
__global__ __launch_bounds__(256) void nrm_k(const float* __restrict__ inV,
                                             const float* __restrict__ inA,
                                             float* __restrict__ outV,
                                             float* __restrict__ outA) {
  const float* in = blockIdx.y ? inA : inV;
  float* out      = blockIdx.y ? outA : outV;
  const int wave = threadIdx.x >> 5, lane = threadIdx.x & 31;
  const int row = blockIdx.x * 8 + wave;
  const float* src = in + (size_t)row * DN;
  float4 r[4];
  float s = 0.f;
#pragma unroll
  for (int i = 0; i < 4; ++i) {
    r[i] = *(const float4*)(src + i * 128 + lane * 4);
    s += r[i].x * r[i].x + r[i].y * r[i].y + r[i].z * r[i].z + r[i].w * r[i].w;
  }
#pragma unroll
  for (int off = 16; off > 0; off >>= 1) s += __shfl_xor(s, off, 32);
  const float scale = 1.0f / fmaxf(sqrtf(s), 1e-12f);
  float* dst = out + (size_t)row * DN;
#pragma unroll
  for (int i = 0; i < 4; ++i) {
    float4 o;
    o.x = r[i].x * scale; o.y = r[i].y * scale;
    o.z = r[i].z * scale; o.w = r[i].w * scale;
    *(float4*)(dst + i * 128 + lane * 4) = o;
  }
}

// ------------------------------------------------- fused GEMM + top-k + LSE
__global__ __launch_bounds__(256) void simtopk_k(const float* __restrict__ Xn,
                                                 const float* __restrict__ Yn,
                                                 float* __restrict__ rloss) {
  __shared__ float Bs[16 * NTILE * BS_STRIDE];   // 32 cols x 512 K (padded)

  const float* X = (blockIdx.y == 0) ? Xn : Yn;  // rows of sim (this dir)
  const float* Y = (blockIdx.y == 0) ? Yn : Xn;  // cols of sim (this dir)
  float* outl    = rloss + (size_t)blockIdx.y * BN;

  const int tid  = threadIdx.x;
  const int wave = tid >> 5, lane = tid & 31;
  const int half = lane >> 4, l15 = lane & 15;
  const int rowBase = blockIdx.x * ROWS_PER_WG + wave * 16;

  // A fragment source (16x4 f32 layout: row = l15, k-offset = 2*half)
  const float* ap  = X + (size_t)(rowBase + l15) * DN + 2 * half;
  const float* bp0 = &Bs[l15 * BS_STRIDE + 2 * half];
  const float* bp1 = &Bs[(16 + l15) * BS_STRIDE + 2 * half];

  float t[8][KTOP];   // per-lane running top-5 for the 8 rows this lane holds
  float pos[8];       // diagonal (positive) value, exactly one lane sees it
#pragma unroll
  for (int j = 0; j < 8; ++j) {
    pos[j] = neg_inf();
#pragma unroll
    for (int i = 0; i < KTOP; ++i) t[j][i] = neg_inf();
  }

  const int NCOLS = 16 * NTILE;                  // 32 columns per pass
  for (int ct = 0; ct < BN / NCOLS; ++ct) {
    __syncthreads();                             // previous panel consumed
    // -- cooperative stage of B panel: 32 cols x 512 K (64 KB), padded rows
    for (int idx = tid; idx < NCOLS * (DN / 4); idx += 256) {
      const int c  = idx >> 7;                   // 0..31 (128 float4 per row)
      const int kq = (idx & 127) << 2;           // 0,4,...,508
      float4 vv = *(const float4*)(Y + (size_t)(ct * NCOLS + c) * DN + kq);
      *(float4*)(&Bs[c * BS_STRIDE + kq]) = vv;
    }
    if (ct + 1 < BN / NCOLS) {                   // global_prefetch next panel
      __builtin_prefetch(Y + (size_t)((ct + 1) * NCOLS) * DN + tid * 64, 0, 1);
    }
    __syncthreads();

    // -- two 16x16 output tiles, K=512: 2 independent WMMA chains,
    //    each A fragment feeds both (halves A-load traffic per WMMA)
    v8f acc0 = {0.f, 0.f, 0.f, 0.f, 0.f, 0.f, 0.f, 0.f};
    v8f acc1 = {0.f, 0.f, 0.f, 0.f, 0.f, 0.f, 0.f, 0.f};
#pragma unroll 4
    for (int k = 0; k < DN; k += 4) {
      v2f a  = *(const v2f*)(ap + k);            // global, L0-hot strip
      v2f b0 = *(const v2f*)(bp0 + k);           // LDS, bank-conflict-free
      v2f b1 = *(const v2f*)(bp1 + k);
      acc0 = __builtin_amdgcn_wmma_f32_16x16x4_f32(
          false, a, false, b0, (short)0, acc0, false, false);
      acc1 = __builtin_amdgcn_wmma_f32_16x16x4_f32(
          false, a, false, b1, (short)0, acc1, false, false);
    }

    // -- branch-free streaming insert into per-row top-5 (+ positive)
#pragma unroll
    for (int nt = 0; nt < NTILE; ++nt) {
      const int col = ct * NCOLS + nt * 16 + l15;
#pragma unroll
      for (int j = 0; j < 8; ++j) {
        const float val = (nt ? acc1[j] : acc0[j]) * INV_T;
        const int row = rowBase + j + 8 * half;
        const bool isdiag = (col == row);
        pos[j] = isdiag ? val : pos[j];
        float x = isdiag ? neg_inf() : val;
#pragma unroll
        for (int i = 0; i < KTOP; ++i) {         // sorted-descending network
          const float mx = fmaxf(t[j][i], x);
          x = fminf(t[j][i], x);
          t[j][i] = mx;
        }
      }
    }
  }

  // -- merge the 16 per-lane partial top-5 lists of each row via shuffles
  const int srcBase = half << 4;
#pragma unroll
  for (int j = 0; j < 8; ++j) {
    float m[KTOP];
    float p = neg_inf();
#pragma unroll
    for (int i = 0; i < KTOP; ++i) m[i] = neg_inf();
    for (int s = 0; s < 16; ++s) {
      p = fmaxf(p, __shfl(pos[j], srcBase + s, 32));
#pragma unroll
      for (int i = 0; i < KTOP; ++i) {
        float x = __shfl(t[j][i], srcBase + s, 32);
#pragma unroll
        for (int q = 0; q < KTOP; ++q) {
          const float mx = fmaxf(m[q], x);
          x = fminf(m[q], x);
          m[q] = mx;
        }
      }
    }
    // logsumexp([pos, top5]) - pos
    float mx = fmaxf(p, m[0]);
    float ssum = expf(p - mx);
#pragma unroll
    for (int i = 0; i < KTOP; ++i) ssum += expf(m[i] - mx);
    const float lse = mx + logf(ssum);
    if (l15 == j) outl[rowBase + j + 8 * half] = lse - p;
  }
}

// ------------------------------------------------------------------- reduce
__global__ __launch_bounds__(256) void red_k(const float* __restrict__ rloss,
                                             float* __restrict__ out) {
  __shared__ float sm[256];
  float s = 0.f;
  for (int i = threadIdx.x; i < 2 * BN; i += 256) s += rloss[i];
  sm[threadIdx.x] = s;
  __syncthreads();
  for (int off = 128; off > 0; off >>= 1) {
    if (threadIdx.x < off) sm[threadIdx.x] += sm[threadIdx.x + off];
    __syncthreads();
  }
  if (threadIdx.x == 0) out[0] = sm[0] * (1.0f / (2.0f * BN));
}

// ------------------------------------------------------------------- launch
extern "C" void kernel_launch(void* const* d_in, const int* in_sizes, int n_in,
                              void* d_out, int out_size, void* d_ws,
                              size_t ws_size, hipStream_t stream) {
  (void)in_sizes; (void)n_in; (void)out_size; (void)ws_size;
  const float* v = (const float*)d_in[0];
  const float* a = (const float*)d_in[1];

  float* Xn = (float*)d_ws;                    // 8192*512 f32 (16 MB)
  float* Yn = Xn + (size_t)BN * DN;            // 8192*512 f32 (16 MB)
  float* rl = Yn + (size_t)BN * DN;            // 2*8192 f32 row losses

  nrm_k<<<dim3(BN / 8, 2), 256, 0, stream>>>(v, a, Xn, Yn);
  simtopk_k<<<dim3(BN / ROWS_PER_WG, 2), 256, 0, stream>>>(Xn, Yn, rl);
  red_k<<<1, 256, 0, stream>>>(rl, (float*)d_out);
}